// NeuralSplineFourierFilter_Activation_4l_55198919688704
// MI455X (gfx1250) — compile-verified
//
#include <hip/hip_runtime.h>

typedef __attribute__((ext_vector_type(2))) float v2f;
typedef __attribute__((ext_vector_type(8))) float v8f;

__device__ __forceinline__ float lrelu(float v) { return v > 0.0f ? v : 0.01f * v; }

struct BranchPtrs {
  const float *l1w, *l1b, *l2w, *l2b, *ww, *wb, *kw, *kb;
  const float *a1w, *a1b, *a2w, *a2b, *a3w, *a3b;
};
struct SetupArgs {
  const float* par;
  BranchPtrs br[4];
  const float *n1w, *n1b, *n2w, *n2b, *n3w, *n3b;
  float* tbl;   // d_ws: 4 branches x 32 floats (knots[14]+pad, ctrl[10]+pad)
  float* tail;  // d_out + 4*N : a1[2] a2[2] a3[2] a4[2] actpars[3]
};

// ---------------------------------------------------------------------------
// Kernel 1: parameter nets. One workgroup, 8 waves.
// 128x128 layers computed as GEMV via V_WMMA_F32_16X16X4_F32.
// ---------------------------------------------------------------------------
__global__ __launch_bounds__(256) void setup_kernel(SetupArgs args) {
  __shared__ float s_par[8];
  __shared__ float s_h[128];
  __shared__ float s_h2[128];
  const int tid = threadIdx.x;

  if (tid < 7) s_par[tid] = args.par[tid];
  if (tid == 7) s_par[7] = 0.0f;
  __syncthreads();

  // ---- knot nets: 7->16->16->{9 weights, 7 knot gaps}; one lane per branch ----
  if (tid < 4) {
    const BranchPtrs P = args.br[tid];
    float n1[16], n2[16];
    for (int j = 0; j < 16; ++j) {
      float acc = P.l1b[j];
      for (int i = 0; i < 7; ++i) acc += s_par[i] * P.l1w[i * 16 + j];
      n1[j] = lrelu(acc);
    }
    for (int j = 0; j < 16; ++j) {
      float acc = P.l2b[j];
      for (int i = 0; i < 16; ++i) acc += n1[i] * P.l2w[i * 16 + j];
      n2[j] = lrelu(acc);
    }
    float wv[9];
    for (int j = 0; j < 9; ++j) {
      float acc = P.wb[j];
      for (int i = 0; i < 16; ++i) acc += n2[i] * P.ww[i * 9 + j];
      wv[j] = acc;
    }
    float kv[7];
    float mx = -1e30f;
    for (int j = 0; j < 7; ++j) {
      float acc = P.kb[j];
      for (int i = 0; i < 16; ++i) acc += n2[i] * P.kw[i * 7 + j];
      kv[j] = acc;
      mx = fmaxf(mx, acc);
    }
    float s = 0.0f;
    for (int j = 0; j < 7; ++j) { kv[j] = __expf(kv[j] - mx); s += kv[j]; }
    const float inv = 1.0f / s;
    float* tb = args.tbl + tid * 32;
    tb[0] = tb[1] = tb[2] = tb[3] = 0.0f;         // 3 pad zeros + k0 = 0
    float cum = 0.0f;
    for (int j = 0; j < 7; ++j) { cum += kv[j] * inv; tb[4 + j] = cum; }
    tb[11] = tb[12] = tb[13] = 1.0f;               // 3 pad ones
    tb[14] = tb[15] = 1.0f;                        // slot padding
    tb[16] = 0.0f;                                 // c[0] = 0
    for (int j = 0; j < 9; ++j) tb[17 + j] = wv[j];
    for (int j = 26; j < 32; ++j) tb[j] = 0.0f;    // slot padding
  }
  __syncthreads();

  // ---- activation nets: 7->128->128->{2,3}; five of them ----
  const float* A1W[5] = {args.br[0].a1w, args.br[1].a1w, args.br[2].a1w, args.br[3].a1w, args.n1w};
  const float* A1B[5] = {args.br[0].a1b, args.br[1].a1b, args.br[2].a1b, args.br[3].a1b, args.n1b};
  const float* A2W[5] = {args.br[0].a2w, args.br[1].a2w, args.br[2].a2w, args.br[3].a2w, args.n2w};
  const float* A2B[5] = {args.br[0].a2b, args.br[1].a2b, args.br[2].a2b, args.br[3].a2b, args.n2b};
  const float* A3W[5] = {args.br[0].a3w, args.br[1].a3w, args.br[2].a3w, args.br[3].a3w, args.n3w};
  const float* A3B[5] = {args.br[0].a3b, args.br[1].a3b, args.br[2].a3b, args.br[3].a3b, args.n3b};

  const int lane = tid & 31;
  const int wid = tid >> 5;  // wave id: owns output columns [16*wid, 16*wid+16)

#pragma unroll
  for (int m = 0; m < 5; ++m) {
    // layer 1: 7 -> 128
    if (tid < 128) {
      float acc = A1B[m][tid];
      for (int i = 0; i < 7; ++i) acc += s_par[i] * A1W[m][i * 128 + tid];
      s_h[tid] = lrelu(acc);
    }
    __syncthreads();

    // layer 2: GEMV 1x128 @ 128x128 via WMMA f32 16x16x4.
    // A frag (16x4): only row M=0 nonzero = hidden vector chunk.
    //   lane 0:  {h[k0], h[k0+1]}   lane 16: {h[k0+2], h[k0+3]}   others zero.
    // B frag (4x16): lanes 0-15 hold rows K=k0,k0+1 at N=lane;
    //   lanes 16-31 hold rows K=k0+2,k0+3 at N=lane-16.
    {
      const float* W2 = A2W[m];
      v8f c = {0.f, 0.f, 0.f, 0.f, 0.f, 0.f, 0.f, 0.f};
      const int col = (wid << 4) + (lane & 15);
      const int kshift = (lane >> 4) << 1;  // 0 for lanes<16, 2 for lanes>=16
      for (int kb = 0; kb < 32; ++kb) {
        const int k0 = kb * 4;
        v2f a;
        a.x = 0.0f; a.y = 0.0f;
        if (lane == 0)  { a.x = s_h[k0 + 0]; a.y = s_h[k0 + 1]; }
        if (lane == 16) { a.x = s_h[k0 + 2]; a.y = s_h[k0 + 3]; }
        v2f b;
        b.x = W2[(k0 + kshift + 0) * 128 + col];
        b.y = W2[(k0 + kshift + 1) * 128 + col];
        c = __builtin_amdgcn_wmma_f32_16x16x4_f32(false, a, false, b, (short)0, c,
                                                  false, false);
      }
      // D row M=0 lives in lanes 0-15, VGPR 0
      if (lane < 16) {
        const float o = c[0] + A2B[m][col];
        s_h2[col] = lrelu(o);
      }
    }
    __syncthreads();

    // layer 3: 128 -> nout, write tail outputs
    const int nout = (m < 4) ? 2 : 3;
    if (tid < nout) {
      float acc = A3B[m][tid];
      for (int i = 0; i < 128; ++i) acc += s_h2[i] * A3W[m][i * nout + tid];
      args.tail[(m < 4 ? 2 * m : 8) + tid] = acc;
    }
    __syncthreads();
  }
}

// ---------------------------------------------------------------------------
// Kernel 2: bandwidth-bound De Boor spline over 16.7M points, 4 branches.
// 4 B in + 16 B out per element -> ~14us floor at 23.3 TB/s.
// ---------------------------------------------------------------------------
__global__ __launch_bounds__(256) void spline_kernel(const float* __restrict__ x,
                                                     const float* __restrict__ tbl,
                                                     float* __restrict__ out, int n) {
  __shared__ float s_tbl[128];
  const int tid = threadIdx.x;
  if (tid < 128) s_tbl[tid] = tbl[tid];
  __syncthreads();

  const int grp = blockIdx.x * 256 + tid;  // group of 4 consecutive elements
  const int ngrp = n >> 2;
  if (grp >= ngrp) return;

  const float4 xv = reinterpret_cast<const float4*>(x)[grp];
  // stream look-ahead (lowers to global_prefetch_b8)
  __builtin_prefetch(reinterpret_cast<const float4*>(x) + grp + 8192, 0, 0);

  const float RS3 = 0.57735026918962576f;  // 1/sqrt(3)
  float xs[4];
  xs[0] = fminf(fmaxf(xv.x * RS3, 0.0f), 0.9999f);
  xs[1] = fminf(fmaxf(xv.y * RS3, 0.0f), 0.9999f);
  xs[2] = fminf(fmaxf(xv.z * RS3, 0.0f), 0.9999f);
  xs[3] = fminf(fmaxf(xv.w * RS3, 0.0f), 0.9999f);

#pragma unroll
  for (int b = 0; b < 4; ++b) {
    const float* t = s_tbl + b * 32;
    const float* c = t + 16;
    const float t4 = t[4], t5 = t[5], t6 = t[6], t7 = t[7], t8 = t[8], t9 = t[9];
    float rr[4];
#pragma unroll
    for (int e = 0; e < 4; ++e) {
      const float v = xs[e];
      const int k = 3 + (v >= t4) + (v >= t5) + (v >= t6) + (v >= t7) + (v >= t8) +
                    (v >= t9);  // k in [3,9]
      float d0 = c[k - 3], d1 = c[k - 2], d2 = c[k - 1], d3 = c[k];
      const float tkm2 = t[k - 2], tkm1 = t[k - 1], tk = t[k];
      const float tk1 = t[k + 1], tk2 = t[k + 2], tk3 = t[k + 3];
      float a;
      // r = 1
      a = (v - tk)   * __builtin_amdgcn_rcpf(tk3 - tk);   d3 = (1.0f - a) * d2 + a * d3;
      a = (v - tkm1) * __builtin_amdgcn_rcpf(tk2 - tkm1); d2 = (1.0f - a) * d1 + a * d2;
      a = (v - tkm2) * __builtin_amdgcn_rcpf(tk1 - tkm2); d1 = (1.0f - a) * d0 + a * d1;
      // r = 2
      a = (v - tk)   * __builtin_amdgcn_rcpf(tk2 - tk);   d3 = (1.0f - a) * d2 + a * d3;
      a = (v - tkm1) * __builtin_amdgcn_rcpf(tk1 - tkm1); d2 = (1.0f - a) * d1 + a * d2;
      // r = 3
      a = (v - tk)   * __builtin_amdgcn_rcpf(tk1 - tk);   d3 = (1.0f - a) * d2 + a * d3;
      rr[e] = d3;
    }
    float4 r;
    r.x = rr[0]; r.y = rr[1]; r.z = rr[2]; r.w = rr[3];
    reinterpret_cast<float4*>(out + (size_t)b * (size_t)n)[grp] = r;
  }
}

// ---------------------------------------------------------------------------
// Host launcher. Detects x (unique largest input) to disambiguate input
// ordering: insertion order (x, par, params...) vs jax-sorted (par, params..., x).
// ---------------------------------------------------------------------------
extern "C" void kernel_launch(void* const* d_in, const int* in_sizes, int n_in,
                              void* d_out, int out_size, void* d_ws, size_t ws_size,
                              hipStream_t stream) {
  int xi = 0;
  for (int i = 1; i < n_in; ++i)
    if (in_sizes[i] > in_sizes[xi]) xi = i;
  const float* xp = (const float*)d_in[xi];
  const int N = in_sizes[xi];

  SetupArgs A;
  float* tbl = (float*)d_ws;
  float* out = (float*)d_out;
  A.tbl = tbl;
  A.tail = out + (size_t)4 * (size_t)N;

  if (xi == 0) {
    // insertion order: x, par, per-branch {l1(w,b), l2(w,b), w(w,b), k(w,b),
    //                                      a1(w,b), a2(w,b), a3(w,b)}, net3{...}
    A.par = (const float*)d_in[1];
    for (int b = 0; b < 4; ++b) {
      const int o = 2 + b * 14;
      A.br[b].l1w = (const float*)d_in[o + 0];  A.br[b].l1b = (const float*)d_in[o + 1];
      A.br[b].l2w = (const float*)d_in[o + 2];  A.br[b].l2b = (const float*)d_in[o + 3];
      A.br[b].ww  = (const float*)d_in[o + 4];  A.br[b].wb  = (const float*)d_in[o + 5];
      A.br[b].kw  = (const float*)d_in[o + 6];  A.br[b].kb  = (const float*)d_in[o + 7];
      A.br[b].a1w = (const float*)d_in[o + 8];  A.br[b].a1b = (const float*)d_in[o + 9];
      A.br[b].a2w = (const float*)d_in[o + 10]; A.br[b].a2b = (const float*)d_in[o + 11];
      A.br[b].a3w = (const float*)d_in[o + 12]; A.br[b].a3b = (const float*)d_in[o + 13];
    }
    const int o = 2 + 56;
    A.n1w = (const float*)d_in[o + 0]; A.n1b = (const float*)d_in[o + 1];
    A.n2w = (const float*)d_in[o + 2]; A.n2b = (const float*)d_in[o + 3];
    A.n3w = (const float*)d_in[o + 4]; A.n3b = (const float*)d_in[o + 5];
  } else {
    // jax-sorted pytree: par, per-branch {a1(b,w), a2(b,w), a3(b,w), k(b,w),
    //                                     l1(b,w), l2(b,w), w(b,w)}, net3, x
    A.par = (const float*)d_in[0];
    for (int b = 0; b < 4; ++b) {
      const int o = 1 + b * 14;
      A.br[b].a1b = (const float*)d_in[o + 0];  A.br[b].a1w = (const float*)d_in[o + 1];
      A.br[b].a2b = (const float*)d_in[o + 2];  A.br[b].a2w = (const float*)d_in[o + 3];
      A.br[b].a3b = (const float*)d_in[o + 4];  A.br[b].a3w = (const float*)d_in[o + 5];
      A.br[b].kb  = (const float*)d_in[o + 6];  A.br[b].kw  = (const float*)d_in[o + 7];
      A.br[b].l1b = (const float*)d_in[o + 8];  A.br[b].l1w = (const float*)d_in[o + 9];
      A.br[b].l2b = (const float*)d_in[o + 10]; A.br[b].l2w = (const float*)d_in[o + 11];
      A.br[b].wb  = (const float*)d_in[o + 12]; A.br[b].ww  = (const float*)d_in[o + 13];
    }
    const int o = 1 + 56;
    A.n1b = (const float*)d_in[o + 0]; A.n1w = (const float*)d_in[o + 1];
    A.n2b = (const float*)d_in[o + 2]; A.n2w = (const float*)d_in[o + 3];
    A.n3b = (const float*)d_in[o + 4]; A.n3w = (const float*)d_in[o + 5];
  }

  hipLaunchKernelGGL(setup_kernel, dim3(1), dim3(256), 0, stream, A);

  const int ngrp = N >> 2;
  const int blocks = (ngrp + 255) / 256;
  hipLaunchKernelGGL(spline_kernel, dim3(blocks), dim3(256), 0, stream, xp, tbl, out, N);
}